// MeshPool_54984171323586
// MI455X (gfx1250) — compile-verified
//
#include <hip/hip_runtime.h>

#define NV   8192      // vertices
#define NE   32768     // edges
#define FD   128       // feature dim
#define TGT  4096      // V//2
#define ROWW 256       // dwords per adjacency row (8192 bits)

typedef float v2f __attribute__((ext_vector_type(2)));
typedef float v8f __attribute__((ext_vector_type(8)));

// ---------------- zero adjacency bitmap (8 MB) ----------------
__global__ __launch_bounds__(256) void mp_zero_kernel(uint4* __restrict__ S4, int n4) {
    int i = blockIdx.x * blockDim.x + threadIdx.x;
    if (i < n4) S4[i] = make_uint4(0u, 0u, 0u, 0u);
}

// ---------------- vertex norms via V_WMMA_F32_16X16X4_F32 ----------------
// Gram tile: D = X(16x4) * X^T(4x16) accumulated over K=128 in steps of 4.
// A and B operand layouts coincide for this Gram computation, so one v2f
// feeds both. Diagonal of the 16x16 accumulator = squared norms.
__global__ __launch_bounds__(256) void mp_norms_kernel(const float* __restrict__ image,
                                                       float* __restrict__ norms) {
    __shared__ float tile[8][256];           // one 16x16 f32 tile per wave
    const int lane = threadIdx.x & 31;
    const int wave = threadIdx.x >> 5;
    const int tileIdx = blockIdx.x * 8 + wave;   // 512 tiles of 16 vertices
    const int base = tileIdx * 16;
    const int m = lane & 15;
    const int hiK = (lane >> 4) ? 2 : 0;         // lanes 16-31 carry K+2,K+3
    const float* rowp = image + (size_t)(base + m) * FD + hiK;

    v8f c = {};
    for (int k0 = 0; k0 < FD; k0 += 4) {
        v2f a;
        a.x = rowp[k0];
        a.y = rowp[k0 + 1];
        c = __builtin_amdgcn_wmma_f32_16x16x4_f32(false, a, false, a,
                                                  (short)0, c, false, false);
    }
    float* w = &tile[wave][0];
#pragma unroll
    for (int r = 0; r < 8; ++r) {
        int M = (lane < 16) ? r : (r + 8);
        int N = lane & 15;
        w[M * 16 + N] = c[r];
    }
    __syncthreads();
    if (lane < 16) norms[base + lane] = w[lane * 16 + lane];
}

// ---------------- edge priorities ----------------
__global__ __launch_bounds__(256) void mp_prio_kernel(const int* __restrict__ e0,
                                                      const int* __restrict__ e1,
                                                      const float* __restrict__ norms,
                                                      float* __restrict__ prio) {
    int i = blockIdx.x * blockDim.x + threadIdx.x;
    if (i < NE) prio[i] = norms[e0[i]] + norms[e1[i]];
}

// ---------------- build symmetric adjacency bitmap ----------------
__global__ __launch_bounds__(256) void mp_build_adj_kernel(const int* __restrict__ e0,
                                                           const int* __restrict__ e1,
                                                           unsigned* __restrict__ S) {
    int i = blockIdx.x * blockDim.x + threadIdx.x;
    if (i < NE) {
        int a = e0[i], b = e1[i];
        atomicOr(&S[(size_t)a * ROWW + (b >> 5)], 1u << (b & 31));
        atomicOr(&S[(size_t)b * ROWW + (a >> 5)], 1u << (a & 31));
    }
}

// ---------------- bitonic argsort in 256 KB LDS (single WG) ----------------
__global__ __launch_bounds__(1024) void mp_sort_kernel(const float* __restrict__ prio,
                                                       const int* __restrict__ e0,
                                                       const int* __restrict__ e1,
                                                       int* __restrict__ v0s,
                                                       int* __restrict__ v1s) {
    extern __shared__ unsigned long long buf[];   // 32768 * 8B = 256 KB
    const int tid = threadIdx.x;
    for (int i = tid; i < NE; i += 1024) {
        unsigned kb = __float_as_uint(prio[i]);   // priorities >= 0: bits monotone
        buf[i] = ((unsigned long long)kb << 32) | (unsigned)i;
    }
    __syncthreads();
    for (int k = 2; k <= NE; k <<= 1) {
        for (int j = k >> 1; j > 0; j >>= 1) {
            for (int i = tid; i < NE; i += 1024) {
                int ij = i ^ j;
                if (ij > i) {
                    unsigned long long a = buf[i], b = buf[ij];
                    bool up = (i & k) == 0;
                    if ((a > b) == up) { buf[i] = b; buf[ij] = a; }
                }
            }
            __syncthreads();
        }
    }
    for (int i = tid; i < NE; i += 1024) {
        int idx = (int)(unsigned)buf[i];
        v0s[i] = e0[idx];
        v1s[i] = e1[idx];
    }
}

// ---------------- sequential greedy decisions (single WG, 256 thr) ----------------
__global__ __launch_bounds__(256) void mp_decide_kernel(const int* __restrict__ v0s,
                                                        const int* __restrict__ v1s,
                                                        const float* __restrict__ vs,
                                                        unsigned* __restrict__ S,
                                                        int* __restrict__ flags,
                                                        float* __restrict__ maskOut) {
    __shared__ unsigned vmaskb[ROWW];
    __shared__ unsigned boundb[ROWW];
    __shared__ int sh_v0, sh_v1, sh_quick, sh_cnt;
    const int tid = threadIdx.x;

    unsigned bb = 0;
    for (int b = 0; b < 32; ++b) {
        int v = (tid << 5) | b;
        float x = vs[v * 2], y = vs[v * 2 + 1];
        bool bd = (x < 0.02f) | (x > 0.98f) | (y < 0.02f) | (y > 0.98f);
        bb |= (unsigned)bd << b;
    }
    boundb[tid] = bb;
    vmaskb[tid] = 0xFFFFFFFFu;
    int vcount = NV;
    __syncthreads();

    for (int t = 0; t < NE; ++t) {
        if (t + 1 < NE) {                  // warm next rows (global_prefetch_b8)
            int n0 = v0s[t + 1], n1 = v1s[t + 1];
            __builtin_prefetch(&S[(size_t)n0 * ROWW + tid], 0, 0);
            __builtin_prefetch(&S[(size_t)n1 * ROWW + tid], 0, 0);
        }
        if (tid == 0) {
            int v0 = v0s[t], v1 = v1s[t];
            sh_v0 = v0; sh_v1 = v1; sh_cnt = 0;
            bool a0 = (vmaskb[v0 >> 5] >> (v0 & 31)) & 1u;
            bool a1 = (vmaskb[v1 >> 5] >> (v1 & 31)) & 1u;
            bool b0 = (boundb[v0 >> 5] >> (v0 & 31)) & 1u;
            bool b1 = (boundb[v1 >> 5] >> (v1 & 31)) & 1u;
            sh_quick = (vcount > TGT) && a0 && a1 && (v0 != v1) && !b0 && !b1;
        }
        __syncthreads();
        const int v0 = sh_v0, v1 = sh_v1;
        const int quick = sh_quick;
        unsigned r0 = 0, r1 = 0;
        if (quick) {
            r0 = S[(size_t)v0 * ROWW + tid];
            r1 = S[(size_t)v1 * ROWW + tid];
            int c = __popc(r0 & r1);
            if (c) atomicAdd(&sh_cnt, c);
        }
        __syncthreads();
        const bool ok = quick && (sh_cnt == 2);
        if (ok) {
            unsigned m = r0 | r1;
            if (tid == (v0 >> 5)) m &= ~(1u << (v0 & 31));
            if (tid == (v1 >> 5)) m &= ~(1u << (v1 & 31));
            S[(size_t)v0 * ROWW + tid] = m;
            S[(size_t)v1 * ROWW + tid] = 0u;
            unsigned rem = r1;                       // old nbr(v1): column fixups
            while (rem) {
                int b = __builtin_ctz(rem); rem &= rem - 1;
                int u = (tid << 5) | b;
                if (u != v0 && u != v1) {
                    unsigned* row = S + (size_t)u * ROWW;
                    int w0 = v0 >> 5, w1 = v1 >> 5;
                    if (w0 == w1) {
                        unsigned d = row[w0];
                        d |= (1u << (v0 & 31));
                        d &= ~(1u << (v1 & 31));
                        row[w0] = d;
                    } else {
                        row[w0] |= (1u << (v0 & 31));
                        row[w1] &= ~(1u << (v1 & 31));
                    }
                }
            }
            if (tid == 0) vmaskb[v1 >> 5] &= ~(1u << (v1 & 31));
            vcount--;                                // ok is block-uniform
        }
        if (tid == 0) flags[t] = ok ? 1 : 0;
        __threadfence();
        __syncthreads();
    }

    unsigned mm = vmaskb[tid];
    for (int b = 0; b < 32; ++b) {
        int v = (tid << 5) | b;
        maskOut[v] = ((mm >> b) & 1u) ? 1.0f : 0.0f;
    }
}

// ---------------- feature copy, sequential merges (per-column), masking ----------------
__global__ __launch_bounds__(256) void mp_featinit_kernel(const float* __restrict__ image,
                                                          float* __restrict__ feat) {
    int i = blockIdx.x * blockDim.x + threadIdx.x;
    if (i < NV * FD) feat[i] = image[i];
}

__global__ __launch_bounds__(128) void mp_merge_kernel(const int* __restrict__ v0s,
                                                       const int* __restrict__ v1s,
                                                       const int* __restrict__ flags,
                                                       float* __restrict__ feat) {
    const int d = threadIdx.x;            // columns are independent: no barriers
    for (int t = 0; t < NE; ++t) {
        if (flags[t]) {
            int v0 = v0s[t], v1 = v1s[t];
            float f0 = feat[(size_t)v0 * FD + d];
            float f1 = feat[(size_t)v1 * FD + d];
            feat[(size_t)v0 * FD + d] = 0.5f * (f0 + f1);
        }
    }
}

__global__ __launch_bounds__(256) void mp_mask_kernel(float* __restrict__ out) {
    int i = blockIdx.x * blockDim.x + threadIdx.x;
    if (i < NV * FD) out[i] *= out[NV * FD + (i >> 7)];
}

// ---------------- host-side orchestration ----------------
extern "C" void kernel_launch(void* const* d_in, const int* in_sizes, int n_in,
                              void* d_out, int out_size, void* d_ws, size_t ws_size,
                              hipStream_t stream) {
    (void)in_sizes; (void)n_in; (void)out_size; (void)ws_size;

    const int*   edges = (const int*)d_in[0];   // (2, NE)
    const float* image = (const float*)d_in[1]; // (NV, FD)
    const float* vs    = (const float*)d_in[2]; // (NV, 2)
    float* out = (float*)d_out;                 // [NV*FD feat][NV mask]

    char* ws = (char*)d_ws;
    unsigned* S  = (unsigned*)ws;                                   // 8 MB bitmap
    size_t off = (size_t)NV * (NV / 8);
    float* norms = (float*)(ws + off);            off += (size_t)NV * 4;
    float* prio  = (float*)(ws + off);            off += (size_t)NE * 4;
    int*   v0s   = (int*)(ws + off);              off += (size_t)NE * 4;
    int*   v1s   = (int*)(ws + off);              off += (size_t)NE * 4;
    int*   flags = (int*)(ws + off);

    const int* e0 = edges;
    const int* e1 = edges + NE;

    int n4 = (NV * (NV / 8)) / 16;   // uint4 count = 524288
    mp_zero_kernel<<<n4 / 256, 256, 0, stream>>>((uint4*)S, n4);
    mp_norms_kernel<<<NV / (16 * 8), 256, 0, stream>>>(image, norms);
    mp_prio_kernel<<<NE / 256, 256, 0, stream>>>(e0, e1, norms, prio);
    mp_build_adj_kernel<<<NE / 256, 256, 0, stream>>>(e0, e1, S);
    mp_sort_kernel<<<1, 1024, NE * sizeof(unsigned long long), stream>>>(prio, e0, e1, v0s, v1s);
    mp_decide_kernel<<<1, 256, 0, stream>>>(v0s, v1s, vs, S, flags, out + (size_t)NV * FD);
    mp_featinit_kernel<<<(NV * FD) / 256, 256, 0, stream>>>(image, out);
    mp_merge_kernel<<<1, 128, 0, stream>>>(v0s, v1s, flags, out);
    mp_mask_kernel<<<(NV * FD) / 256, 256, 0, stream>>>(out);
}